// HGNNModel_19327352832289
// MI455X (gfx1250) — compile-verified
//
#include <hip/hip_runtime.h>

// ---------------------------------------------------------------------------
// HGNN (hypergraph conv) for MI455X / gfx1250.
//   u = leaky(emb[:NU] @ fc_u_w^T + b) ; i = leaky(emb[NU:] @ fc_i_w^T + b)
//   per layer: hu = leaky(LN(A @ LN((A^T u) @ Wu^T)));  u += hu; acc += hu
//              hi = leaky(LN(A^T @ LN((A i) @ Wi^T)));  i += hi; acc += hi
//   out = acc / 3
// Dense GEMMs use V_WMMA_F32_16X16X4_F32 (full f32 precision).
// SpMMs are wave-per-edge atomic scatters (L2-resident operands).
// ---------------------------------------------------------------------------

namespace {
constexpr int   kNU   = 100000;
constexpr int   kNI   = 50000;
constexpr int   kDIN  = 64;
constexpr int   kDH   = 128;
constexpr int   kL    = 2;
constexpr int   kNNZ  = 5000000;
constexpr float kSlope = 0.2f;
constexpr float kEps   = 1e-5f;
}

typedef float v2f __attribute__((ext_vector_type(2)));
typedef float v8f __attribute__((ext_vector_type(8)));

__device__ __forceinline__ float leaky_(float x) { return x > 0.f ? x : kSlope * x; }

// ---------------------------------------------------------------------------
__global__ __launch_bounds__(256) void zero_kernel(float4* __restrict__ p, long n4) {
  long i = (long)blockIdx.x * blockDim.x + threadIdx.x;
  long stride = (long)gridDim.x * blockDim.x;
  float4 z; z.x = z.y = z.z = z.w = 0.f;
  for (; i < n4; i += stride) p[i] = z;
}

__global__ __launch_bounds__(256) void scale_kernel(float4* __restrict__ p, long n4, float sc) {
  long i = (long)blockIdx.x * blockDim.x + threadIdx.x;
  long stride = (long)gridDim.x * blockDim.x;
  for (; i < n4; i += stride) {
    float4 t = p[i];
    t.x *= sc; t.y *= sc; t.z *= sc; t.w *= sc;
    p[i] = t;
  }
}

// ---------------------------------------------------------------------------
// fc: one wave computes a 16-row panel of leaky(emb @ W^T + b).
// A tile 16x4 f32 (v2f/lane), B tile 4x16 = W^T slice (v2f/lane), C 16x16 (v8f).
// Writes u or i, and initializes acc (= d_out) with the same values.
__global__ __launch_bounds__(256) void fc_wmma_kernel(
    const float* __restrict__ emb,
    const float* __restrict__ wu, const float* __restrict__ bu,
    const float* __restrict__ wi, const float* __restrict__ bi,
    float* __restrict__ u, float* __restrict__ iv, float* __restrict__ acc) {
  int wid  = (int)((blockIdx.x * blockDim.x + threadIdx.x) >> 5);
  int lane = threadIdx.x & 31;
  if (wid >= (kNU + kNI) / 16) return;
  int r0 = wid * 16;
  bool isU = (r0 < kNU);
  const float* W = isU ? wu : wi;
  const float* B = isU ? bu : bi;
  float* dst = isU ? (u + (long)r0 * kDH) : (iv + (long)(r0 - kNU) * kDH);

  int half = lane >> 4, lm = lane & 15;
  const float* arow = emb + (long)(r0 + lm) * kDIN;

  v8f acc8[8] = {};
  for (int k = 0; k < kDIN; k += 4) {
    v2f a = *(const v2f*)(arow + k + 2 * half);
#pragma unroll
    for (int t = 0; t < 8; ++t) {
      v2f bb = *(const v2f*)(W + (long)(t * 16 + lm) * kDIN + k + 2 * half);
      acc8[t] = __builtin_amdgcn_wmma_f32_16x16x4_f32(
          false, a, false, bb, (short)0, acc8[t], false, false);
    }
  }
#pragma unroll
  for (int t = 0; t < 8; ++t) {
#pragma unroll
    for (int v = 0; v < 8; ++v) {
      int row = half * 8 + v;
      int col = t * 16 + lm;
      float y = leaky_(acc8[t][v] + B[col]);
      dst[(long)row * kDH + col] = y;
      acc[(long)(r0 + row) * kDH + col] = y;
    }
  }
}

// ---------------------------------------------------------------------------
// out[dst[e]] += vals[e] * x[src[e]]   — wave per edge, lane handles 4 columns.
__global__ __launch_bounds__(256) void spmm_kernel(
    const int* __restrict__ src, const int* __restrict__ dst,
    const float* __restrict__ vals, const float* __restrict__ x,
    float* __restrict__ out, int nnz) {
  int e = (int)((blockIdx.x * blockDim.x + threadIdx.x) >> 5);
  int lane = threadIdx.x & 31;
  if (e >= nnz) return;
  int s = src[e], d = dst[e];
  float v = vals[e];
  float4 t = ((const float4*)(x + (long)s * kDH))[lane];
  float* o = out + (long)d * kDH + lane * 4;
  atomicAdd(o + 0, v * t.x);
  atomicAdd(o + 1, v * t.y);
  atomicAdd(o + 2, v * t.z);
  atomicAdd(o + 3, v * t.w);
}

// ---------------------------------------------------------------------------
// out = LayerNorm(X @ W^T, g, b). Wave per 16-row panel; K=128 via WMMA f32.
// LayerNorm done in-register on the WMMA C layout: row m = (lane-half)*8+v
// lives across the 16 lanes of one half (8 tiles, component v). Reduce with
// shfl_xor over masks 1,2,4,8 (stays within the 16-lane half).
__global__ __launch_bounds__(256) void gemm_ln_kernel(
    const float* __restrict__ X, const float* __restrict__ W,
    const float* __restrict__ g, const float* __restrict__ b,
    float* __restrict__ out, int npanels) {
  int wid  = (int)((blockIdx.x * blockDim.x + threadIdx.x) >> 5);
  int lane = threadIdx.x & 31;
  if (wid >= npanels) return;
  int r0 = wid * 16;
  int half = lane >> 4, lm = lane & 15;
  const float* arow = X + (long)(r0 + lm) * kDH;

  v8f acc8[8] = {};
  for (int k = 0; k < kDH; k += 4) {
    v2f a = *(const v2f*)(arow + k + 2 * half);
#pragma unroll
    for (int t = 0; t < 8; ++t) {
      v2f bb = *(const v2f*)(W + (long)(t * 16 + lm) * kDH + k + 2 * half);
      acc8[t] = __builtin_amdgcn_wmma_f32_16x16x4_f32(
          false, a, false, bb, (short)0, acc8[t], false, false);
    }
  }
#pragma unroll
  for (int v = 0; v < 8; ++v) {
    float s = 0.f, q = 0.f;
#pragma unroll
    for (int t = 0; t < 8; ++t) { float x = acc8[t][v]; s += x; q += x * x; }
#pragma unroll
    for (int off = 1; off < 16; off <<= 1) {
      s += __shfl_xor(s, off, 32);
      q += __shfl_xor(q, off, 32);
    }
    float mu   = s * (1.f / kDH);
    float var  = q * (1.f / kDH) - mu * mu;
    float rstd = rsqrtf(var + kEps);
    int row = r0 + half * 8 + v;
#pragma unroll
    for (int t = 0; t < 8; ++t) {
      int col = t * 16 + lm;
      out[(long)row * kDH + col] = (acc8[t][v] - mu) * rstd * g[col] + b[col];
    }
  }
}

// ---------------------------------------------------------------------------
// h = leaky(LN(y2)); xio += h; accp += h.   Wave per row, float4 per lane.
__global__ __launch_bounds__(256) void ln_leaky_add_kernel(
    const float* __restrict__ y2, const float* __restrict__ g,
    const float* __restrict__ b, float* __restrict__ xio,
    float* __restrict__ accp, int nrows) {
  int row  = (int)((blockIdx.x * blockDim.x + threadIdx.x) >> 5);
  int lane = threadIdx.x & 31;
  if (row >= nrows) return;
  float4 t = ((const float4*)(y2 + (long)row * kDH))[lane];
  float s = t.x + t.y + t.z + t.w;
  float q = t.x * t.x + t.y * t.y + t.z * t.z + t.w * t.w;
#pragma unroll
  for (int off = 1; off < 32; off <<= 1) {
    s += __shfl_xor(s, off, 32);
    q += __shfl_xor(q, off, 32);
  }
  float mu   = s * (1.f / kDH);
  float var  = q * (1.f / kDH) - mu * mu;
  float rstd = rsqrtf(var + kEps);
  int c0 = lane * 4;
  float h0 = leaky_((t.x - mu) * rstd * g[c0 + 0] + b[c0 + 0]);
  float h1 = leaky_((t.y - mu) * rstd * g[c0 + 1] + b[c0 + 1]);
  float h2 = leaky_((t.z - mu) * rstd * g[c0 + 2] + b[c0 + 2]);
  float h3 = leaky_((t.w - mu) * rstd * g[c0 + 3] + b[c0 + 3]);

  float4* pu = (float4*)(xio + (long)row * kDH) + lane;
  float4 uo = *pu;
  uo.x += h0; uo.y += h1; uo.z += h2; uo.w += h3;
  *pu = uo;
  float4* pa = (float4*)(accp + (long)row * kDH) + lane;
  float4 ao = *pa;
  ao.x += h0; ao.y += h1; ao.z += h2; ao.w += h3;
  *pa = ao;
}

// ---------------------------------------------------------------------------
extern "C" void kernel_launch(void* const* d_in, const int* in_sizes, int n_in,
                              void* d_out, int out_size, void* d_ws, size_t ws_size,
                              hipStream_t stream) {
  const float* emb    = (const float*)d_in[0];
  const float* fcu_w  = (const float*)d_in[1];
  const float* fcu_b  = (const float*)d_in[2];
  const float* fci_w  = (const float*)d_in[3];
  const float* fci_b  = (const float*)d_in[4];
  const float* vals   = (const float*)d_in[5];
  const float* Wu     = (const float*)d_in[6];
  const float* ln1g_u = (const float*)d_in[7];
  const float* ln1b_u = (const float*)d_in[8];
  const float* ln2g_u = (const float*)d_in[9];
  const float* ln2b_u = (const float*)d_in[10];
  const float* Wi     = (const float*)d_in[11];
  const float* ln1g_i = (const float*)d_in[12];
  const float* ln1b_i = (const float*)d_in[13];
  const float* ln2g_i = (const float*)d_in[14];
  const float* ln2b_i = (const float*)d_in[15];
  const int*   rows   = (const int*)d_in[16];
  const int*   cols   = (const int*)d_in[17];

  float* acc = (float*)d_out;                  // [NU+NI, 128] running sum
  float* u   = (float*)d_ws;                   // [NU, 128]
  float* iv  = u  + (long)kNU * kDH;           // [NI, 128]
  float* t1  = iv + (long)kNI * kDH;           // [NU, 128] scratch
  float* t2  = t1 + (long)kNU * kDH;           // [NU, 128] scratch

  const int spmm_grid = (kNNZ * 32 + 255) / 256;

  // fc + acc init
  {
    int panels = (kNU + kNI) / 16;  // 9375
    fc_wmma_kernel<<<(panels + 7) / 8, 256, 0, stream>>>(
        emb, fcu_w, fcu_b, fci_w, fci_b, u, iv, acc);
  }

  for (int k = 0; k < kL; ++k) {
    const float* Wuk = Wu + (long)k * kDH * kDH;
    const float* Wik = Wi + (long)k * kDH * kDH;

    // ---- user branch: hu = leaky(LN2(A @ LN1((A^T u) @ Wu^T))) ----
    zero_kernel<<<2048, 256, 0, stream>>>((float4*)t1, (long)kNI * kDH / 4);
    spmm_kernel<<<spmm_grid, 256, 0, stream>>>(rows, cols, vals, u, t1, kNNZ);
    gemm_ln_kernel<<<(kNI / 16 + 7) / 8, 256, 0, stream>>>(
        t1, Wuk, ln1g_u + k * kDH, ln1b_u + k * kDH, t2, kNI / 16);
    zero_kernel<<<2048, 256, 0, stream>>>((float4*)t1, (long)kNU * kDH / 4);
    spmm_kernel<<<spmm_grid, 256, 0, stream>>>(cols, rows, vals, t2, t1, kNNZ);
    ln_leaky_add_kernel<<<(kNU + 7) / 8, 256, 0, stream>>>(
        t1, ln2g_u + k * kDH, ln2b_u + k * kDH, u, acc, kNU);

    // ---- item branch: hi = leaky(LN2(A^T @ LN1((A i) @ Wi^T))) ----
    zero_kernel<<<2048, 256, 0, stream>>>((float4*)t1, (long)kNU * kDH / 4);
    spmm_kernel<<<spmm_grid, 256, 0, stream>>>(cols, rows, vals, iv, t1, kNNZ);
    gemm_ln_kernel<<<(kNU / 16 + 7) / 8, 256, 0, stream>>>(
        t1, Wik, ln1g_i + k * kDH, ln1b_i + k * kDH, t2, kNU / 16);
    zero_kernel<<<2048, 256, 0, stream>>>((float4*)t1, (long)kNI * kDH / 4);
    spmm_kernel<<<spmm_grid, 256, 0, stream>>>(rows, cols, vals, t2, t1, kNNZ);
    ln_leaky_add_kernel<<<(kNI + 7) / 8, 256, 0, stream>>>(
        t1, ln2g_i + k * kDH, ln2b_i + k * kDH, iv, acc + (long)kNU * kDH, kNI);
  }

  // out = acc / (L+1)
  scale_kernel<<<2048, 256, 0, stream>>>(
      (float4*)acc, (long)(kNU + kNI) * kDH / 4, 1.f / 3.f);
}